// PScan_21397527069323
// MI455X (gfx1250) — compile-verified
//
#include <hip/hip_runtime.h>
#include <stdint.h>

// Problem constants (from reference): B=4, L=1024, C=64, R=16
#define PS_L       1024
#define PS_W       16384    // C*R*R  : elements per (b, l)
#define PS_CR      1024     // C*R    : A elements per (b, l)
#define PS_BLOCK   128
#define PS_STAGES  24       // async pipeline depth (2 async ops per stage)

typedef float v4f __attribute__((ext_vector_type(4)));

__global__ __launch_bounds__(PS_BLOCK) void
pscan_diag_async_kernel(const float* __restrict__ A,
                        const float* __restrict__ X,
                        float* __restrict__ Y)
{
    // Per-block staging: 24 stages * 128 threads * (16B X + 4B A) = 60 KB
    __shared__ v4f   xstage[PS_STAGES][PS_BLOCK];
    __shared__ float astage[PS_STAGES][PS_BLOCK];

    const int tid = threadIdx.x;
    const int t   = blockIdx.x * PS_BLOCK + tid;   // [0, 16384)
    const int b   = t >> 12;                       // batch index (t / 4096)
    const int e4  = t & 4095;                      // float4 slot within (b, l)

    const float* __restrict__ xp = X + (size_t)b * PS_L * PS_W  + (size_t)e4 * 4;
    const float* __restrict__ ap = A + (size_t)b * PS_L * PS_CR + (size_t)(e4 >> 2);
    float*       __restrict__ yc = Y + (size_t)b * PS_L * PS_W  + (size_t)e4 * 4;

    // Wave-relative LDS byte offsets (flat->LDS uses addr[31:0], ISA 10.2)
    const uint32_t ldsx0 = (uint32_t)(uintptr_t)&xstage[0][tid];
    const uint32_t ldsa0 = (uint32_t)(uintptr_t)&astage[0][tid];

    uint64_t gx = (uint64_t)(uintptr_t)xp;
    uint64_t ga = (uint64_t)(uintptr_t)ap;

    // ---- Preamble: fill the PS_STAGES-deep async pipeline ----
#pragma unroll
    for (int s = 0; s < PS_STAGES; ++s) {
        uint32_t lx = ldsx0 + (uint32_t)s * (PS_BLOCK * 16);
        uint32_t la = ldsa0 + (uint32_t)s * (PS_BLOCK * 4);
        asm volatile("global_load_async_to_lds_b128 %0, %1, off"
                     :: "v"(lx), "v"(gx) : "memory");
        asm volatile("global_load_async_to_lds_b32 %0, %1, off"
                     :: "v"(la), "v"(ga) : "memory");
        gx += (uint64_t)PS_W  * 4u;
        ga += (uint64_t)PS_CR * 4u;
    }

    // Prologue: pull stage 0 into registers (issued 48 ops; stage 0 done at <=46)
    asm volatile("s_wait_asynccnt 46" ::: "memory");
    v4f   xr = xstage[0][tid];
    float ar = astage[0][tid];

    v4f y = (v4f){0.f, 0.f, 0.f, 0.f};
    int s_pref = 1;   // slot holding stage l+1: ds_read this iteration, use next
    int s_fill = 0;   // slot whose data the FMAs below consume: refill after use

    // ---- Steady state ----
    // issued = 2(S+l); stage l+1 complete  <=>  asynccnt <= 2S-4 = 44
    for (int l = 0; l < PS_L - PS_STAGES; ++l) {
        asm volatile("s_wait_asynccnt 44" ::: "memory");
        v4f   xn = xstage[s_pref][tid];    // prefetch stage l+1 (consumed next iter)
        float an = astage[s_pref][tid];

        // consume stage l (read last iteration; compiler's dscnt wait guards it)
        y.x = fmaf(ar, y.x, xr.x);
        y.y = fmaf(ar, y.y, xr.y);
        y.z = fmaf(ar, y.z, xr.z);
        y.w = fmaf(ar, y.w, xr.w);
        __builtin_nontemporal_store(y, (v4f*)yc);   // anchors FMA+dscnt-wait before refill
        yc += PS_W;

        // refill the slot just consumed with stage l+S (ds_read already complete)
        {
            uint32_t lx = ldsx0 + (uint32_t)s_fill * (PS_BLOCK * 16);
            uint32_t la = ldsa0 + (uint32_t)s_fill * (PS_BLOCK * 4);
            asm volatile("global_load_async_to_lds_b128 %0, %1, off"
                         :: "v"(lx), "v"(gx) : "memory");
            asm volatile("global_load_async_to_lds_b32 %0, %1, off"
                         :: "v"(la), "v"(ga) : "memory");
            gx += (uint64_t)PS_W  * 4u;
            ga += (uint64_t)PS_CR * 4u;
        }

        xr = xn; ar = an;
        s_pref = (s_pref + 1 == PS_STAGES) ? 0 : s_pref + 1;
        s_fill = (s_fill + 1 == PS_STAGES) ? 0 : s_fill + 1;
    }

    // ---- Tail: nothing left to issue; drain with full waits ----
    for (int l = PS_L - PS_STAGES; l < PS_L - 1; ++l) {
        asm volatile("s_wait_asynccnt 0" ::: "memory");
        v4f   xn = xstage[s_pref][tid];
        float an = astage[s_pref][tid];
        y.x = fmaf(ar, y.x, xr.x);
        y.y = fmaf(ar, y.y, xr.y);
        y.z = fmaf(ar, y.z, xr.z);
        y.w = fmaf(ar, y.w, xr.w);
        __builtin_nontemporal_store(y, (v4f*)yc);
        yc += PS_W;
        xr = xn; ar = an;
        s_pref = (s_pref + 1 == PS_STAGES) ? 0 : s_pref + 1;
    }

    // ---- Final step (l = L-1): registers already hold the last stage ----
    y.x = fmaf(ar, y.x, xr.x);
    y.y = fmaf(ar, y.y, xr.y);
    y.z = fmaf(ar, y.z, xr.z);
    y.w = fmaf(ar, y.w, xr.w);
    __builtin_nontemporal_store(y, (v4f*)yc);
}

extern "C" void kernel_launch(void* const* d_in, const int* in_sizes, int n_in,
                              void* d_out, int out_size, void* d_ws, size_t ws_size,
                              hipStream_t stream) {
    (void)in_sizes; (void)n_in; (void)out_size; (void)d_ws; (void)ws_size;
    const float* A = (const float*)d_in[0];   // [B, L, C, R]
    const float* X = (const float*)d_in[1];   // [B, L, C, R, R]
    float*       Y = (float*)d_out;           // [B, L, C, R, R]

    const int total_threads = 4 * 4096;       // B * (C*R*R/4) = 16384
    dim3 grid(total_threads / PS_BLOCK);      // 128 blocks
    dim3 block(PS_BLOCK);                     // 128 threads = 4 waves
    pscan_diag_async_kernel<<<grid, block, 0, stream>>>(A, X, Y);
}